// CensoredLoss_Sub_73023033967188
// MI455X (gfx1250) — compile-verified
//
#include <hip/hip_runtime.h>

typedef float v2f __attribute__((ext_vector_type(2)));
typedef float v4f __attribute__((ext_vector_type(4)));
typedef float v8f __attribute__((ext_vector_type(8)));

#define CL_EPS 1e-8f

// ---------------------------------------------------------------------------
// Kernel 1: grid-stride streaming pass. One iteration = 2 positions:
//   global_load_b128 (outputs), global_load_b128 (targets), global_load_b64
//   (weights), all non-temporal (single-use stream; 160 MiB total, HBM-bound).
// Per-lane accumulators reduced per-wave with V_WMMA_F32_16X16X4_F32.
// ---------------------------------------------------------------------------
__global__ __launch_bounds__(256) void censored_partial_kernel(
    const float* __restrict__ outputs,
    const float* __restrict__ targets,
    const float* __restrict__ weights,
    float* __restrict__ partial,       // [gridDim.x * 2]
    long long nChunks)                 // chunks of 2 positions
{
    const v4f* __restrict__ o4 = (const v4f*)outputs;
    const v4f* __restrict__ t4 = (const v4f*)targets;
    const v2f* __restrict__ w2 = (const v2f*)weights;

    const long long tid    = (long long)blockIdx.x * blockDim.x + threadIdx.x;
    const long long stride = (long long)gridDim.x * blockDim.x;

    float acc0 = 0.0f;   // channel 0 partial
    float acc1 = 0.0f;   // channel 1 partial

    for (long long i = tid; i < nChunks; i += stride) {
        if (i + stride < nChunks) {
            // speculative prefetch one stride ahead -> global_prefetch_b8
            __builtin_prefetch(&o4[i + stride], 0, 0);
            __builtin_prefetch(&t4[i + stride], 0, 0);
        }
        v4f o = __builtin_nontemporal_load(&o4[i]);   // o0a o1a o0b o1b
        v4f t = __builtin_nontemporal_load(&t4[i]);   // t0a t1a t0b t1b
        v2f w = __builtin_nontemporal_load(&w2[i]);   // wa  wb

        // position a
        float la0 = __logf((1.0f - o.x) + CL_EPS);
        float la1 = __logf(o.x + CL_EPS);
        float lb0 = __logf((1.0f - o.y) + CL_EPS);
        float lb1 = __logf(o.y + CL_EPS);
        acc0 = __fmaf_rn((t.x * la0 + t.y * la1), w.x, acc0);
        acc1 = __fmaf_rn((t.x * lb0 + t.y * lb1), w.x, acc1);

        // position b
        float ma0 = __logf((1.0f - o.z) + CL_EPS);
        float ma1 = __logf(o.z + CL_EPS);
        float mb0 = __logf((1.0f - o.w) + CL_EPS);
        float mb1 = __logf(o.w + CL_EPS);
        acc0 = __fmaf_rn((t.z * ma0 + t.w * ma1), w.y, acc0);
        acc1 = __fmaf_rn((t.z * mb0 + t.w * mb1), w.y, acc1);
    }

    // ---- wave32 reduction via V_WMMA_F32_16X16X4_F32 ----
    // A (16x4 f32, 2 VGPRs): VGPR0 = acc0 -> K=0 (lanes 0-15) / K=2 (16-31)
    //                        VGPR1 = acc1 -> K=1 (lanes 0-15) / K=3 (16-31)
    // So row m: [acc0_m, acc1_m, acc0_{m+16}, acc1_{m+16}].
    // B (4x16): column 0 selects K={0,2} (channel 0), column 1 selects K={1,3}.
    const unsigned lane = threadIdx.x & 31u;
    const unsigned n    = lane & 15u;

    v2f a; a.x = acc0; a.y = acc1;
    v2f b; b.x = (n == 0u) ? 1.0f : 0.0f;   // rows K0/K2
           b.y = (n == 1u) ? 1.0f : 0.0f;   // rows K1/K3
    v8f c = {};
    c = __builtin_amdgcn_wmma_f32_16x16x4_f32(
            /*neg_a=*/false, a, /*neg_b=*/false, b,
            /*c_mod=*/(short)0, c, /*reuse_a=*/false, /*reuse_b=*/false);

    // D[m][n]: lane (n) VGPR j holds M=j (lanes 0-15) / M=j+8 (lanes 16-31).
    // Summing the 8 D regs gives half-row-range sums; shfl_xor(16) completes.
    float s = c[0] + c[1] + c[2] + c[3] + c[4] + c[5] + c[6] + c[7];
    s += __shfl_xor(s, 16, 32);
    // now lane 0 holds wave-sum of channel 0, lane 1 holds channel 1

    __shared__ float wsum[8][2];
    const unsigned wave = threadIdx.x >> 5;
    if (lane < 2u) wsum[wave][lane] = s;
    __syncthreads();

    if (threadIdx.x < 2u) {
        float tsum = 0.0f;
        #pragma unroll
        for (int wv = 0; wv < 8; ++wv) tsum += wsum[wv][threadIdx.x];
        partial[(long long)blockIdx.x * 2 + threadIdx.x] = tsum;
    }
}

// ---------------------------------------------------------------------------
// Kernel 2: deterministic final combine (fixed tree order, no float atomics).
// Even LDS slots carry channel 0, odd slots channel 1.
// ---------------------------------------------------------------------------
__global__ __launch_bounds__(256) void censored_final_kernel(
    const float* __restrict__ partial, float* __restrict__ out,
    int nBlocks, float scale)
{
    __shared__ float lds[256];
    const int tid = threadIdx.x;
    const int ch  = tid & 1;

    float s = 0.0f;
    for (int i = (tid >> 1); i < nBlocks; i += 128)
        s += partial[i * 2 + ch];
    lds[tid] = s;
    __syncthreads();

    for (int st = 128; st >= 2; st >>= 1) {
        if (tid < st) lds[tid] += lds[tid + st];   // channel parity preserved
        __syncthreads();
    }
    if (tid < 2) out[tid] = lds[tid] * scale;      // scale = -1/(B*T)
}

// ---------------------------------------------------------------------------
extern "C" void kernel_launch(void* const* d_in, const int* in_sizes, int n_in,
                              void* d_out, int out_size, void* d_ws, size_t ws_size,
                              hipStream_t stream)
{
    const float* outputs = (const float*)d_in[0];  // (B,T,2) f32
    const float* targets = (const float*)d_in[1];  // (B,T,2) f32
    const float* weights = (const float*)d_in[2];  // (B,T)   f32
    float* out = (float*)d_out;                    // 2 floats
    float* ws  = (float*)d_ws;                     // block partials

    const long long totalPos = (long long)in_sizes[2];   // B*T
    const long long nChunks  = totalPos / 2;             // 2 positions / chunk

    int blocks = 2048;
    const long long maxBlocks = (long long)(ws_size / (2 * sizeof(float)));
    if ((long long)blocks > maxBlocks) blocks = (int)maxBlocks;
    if (blocks < 1) blocks = 1;

    censored_partial_kernel<<<blocks, 256, 0, stream>>>(
        outputs, targets, weights, ws, nChunks);

    const float scale = -1.0f / (float)totalPos;
    censored_final_kernel<<<1, 256, 0, stream>>>(ws, out, blocks, scale);
}